// VSGCLayer_38019050505083
// MI455X (gfx1250) — compile-verified
//
#include <hip/hip_runtime.h>
#include <hip/hip_bf16.h>

// VSGC propagation for MI455X (gfx1250, wave32).
//   out = 0.1*agg*bef_A + 0.9*h + 0.1*ini_h*bef_X,  agg = segment_sum((h*aft_A)[src], dst)
//
// Kernel 1 (init): out = diag(0.1*bef_X) x ini_h + 0.9*h via V_WMMA_F32_16X16X4_F32
//                  (diagonal GEMM -> exact f32 result, XDL units, free vs VALU).
// Kernel 2 (edges): half-wave per edge, float4 per lane, L2-resident gathers +
//                   global_atomic_add_f32 scatter, global_prefetch_b8 pipelining.

typedef __attribute__((ext_vector_type(2))) float v2f;
typedef __attribute__((ext_vector_type(8))) float v8f;

#define ALP 0.1f
#define ONE_MINUS_ALP 0.9f

// ---------------------------------------------------------------------------
// Kernel 1: one wave per 16-node slab. For each 16x16 tile of the [N,64]
// feature matrix:  D = A x B + C  with A = diag(0.1*bef_X[16 rows]) built as
// four 16x4 K-chunks, B = ini_h tile (4x16 chunks), C = 0.9*h tile.
//
// VGPR layouts per CDNA5 ISA 7.12.2 (wave32):
//   A 16x4 f32 : lanes 0-15 M=0..15 (VGPR0:K=0, VGPR1:K=1); lanes 16-31 K=2/3
//   B 4x16 f32 : lanes 0-15 N=0..15 (VGPR0:K=0, VGPR1:K=1); lanes 16-31 K=2/3
//   C/D 16x16  : VGPR r = row r (lanes 0-15) / row r+8 (lanes 16-31)
// ---------------------------------------------------------------------------
__global__ __launch_bounds__(256) void vsgc_init_wmma(
    const float* __restrict__ h, const float* __restrict__ ini_h,
    const float* __restrict__ bef_X, float* __restrict__ out, int n_nodes)
{
    const int lane = threadIdx.x & 31;
    const int wave = (int)((blockIdx.x * blockDim.x + threadIdx.x) >> 5);
    const int base = wave << 4;                 // first node of this 16-row slab
    if (base >= n_nodes) return;                // wave-uniform exit

    if (base + 16 <= n_nodes) {
        // Full tile path: EXEC is all ones (wave-uniform branch) as WMMA requires.
        const int  mn = lane & 15;              // A-row (lanes) == B/C column (lanes)
        const bool hi = lane >= 16;
        const int  kA = hi ? 2 : 0;             // K held in A/B VGPR0 for this half-wave
        const int  rC = hi ? 8 : 0;             // row offset of C/D VGPR0
        const float sX = ALP * bef_X[base + mn];

        const float* hrow = h     + (size_t)base * 64;
        const float* irow = ini_h + (size_t)base * 64;
        float*       orow = out   + (size_t)base * 64;

        #pragma unroll
        for (int ct = 0; ct < 4; ++ct) {        // four 16-wide column tiles of D_FEAT=64
            const int n0 = ct * 16 + mn;

            v8f acc;                            // C = 0.9 * h tile
            #pragma unroll
            for (int r = 0; r < 8; ++r)
                acc[r] = ONE_MINUS_ALP * hrow[(size_t)(r + rC) * 64 + n0];

            #pragma unroll
            for (int c = 0; c < 4; ++c) {       // K-chunks of the diagonal GEMM
                v2f a, b;
                a[0] = (mn == 4 * c + kA)     ? sX : 0.0f;
                a[1] = (mn == 4 * c + kA + 1) ? sX : 0.0f;
                b[0] = irow[(size_t)(4 * c + kA)     * 64 + n0];
                b[1] = irow[(size_t)(4 * c + kA + 1) * 64 + n0];
                // (neg_a, A, neg_b, B, c_mod, C, reuse_a, reuse_b)
                acc = __builtin_amdgcn_wmma_f32_16x16x4_f32(
                    false, a, false, b, (short)0, acc, false, false);
            }

            #pragma unroll
            for (int r = 0; r < 8; ++r)
                orow[(size_t)(r + rC) * 64 + n0] = acc[r];
        }
    } else {
        // Ragged tail (N not a multiple of 16): plain VALU fallback.
        for (int r = base; r < n_nodes; ++r) {
            const float sX = ALP * bef_X[r];
            for (int c = lane; c < 64; c += 32) {
                const size_t idx = (size_t)r * 64 + c;
                out[idx] = ONE_MINUS_ALP * h[idx] + sX * ini_h[idx];
            }
        }
    }
}

// ---------------------------------------------------------------------------
// Kernel 2: half-wave (16 lanes) per edge. Lane q moves float4 q of the 64-f32
// row (16 lanes x 16B = 256B, fully coalesced). Gather source h and the atomic
// destination out are both L2-resident (25.6 MB each vs 192 MB L2).
// Next iteration's row is prefetched (global_prefetch_b8) to hide the random-
// gather latency on top of occupancy.
// ---------------------------------------------------------------------------
__global__ __launch_bounds__(256) void vsgc_edge_scatter(
    const float* __restrict__ h, const float* __restrict__ aft_A,
    const float* __restrict__ bef_A, const int* __restrict__ src,
    const int* __restrict__ dst, float* __restrict__ out, int n_edges)
{
    const int lane  = threadIdx.x & 31;
    const int half  = lane >> 4;                // which half-wave
    const int q     = lane & 15;                // float4 slot within the row
    const int wave  = (int)((blockIdx.x * blockDim.x + threadIdx.x) >> 5);
    const int nHalf = (int)((gridDim.x * blockDim.x) >> 5) * 2;

    for (int e = wave * 2 + half; e < n_edges; e += nHalf) {
        const int s = src[e];                   // broadcast within half-wave
        const int d = dst[e];
        const float coef = ALP * aft_A[s] * bef_A[d];

        const int en = e + nHalf;               // software-pipeline the gather
        if (en < n_edges) {
            const int sn = src[en];
            __builtin_prefetch(h + (size_t)sn * 64 + q * 4, 0, 0);
        }

        const float4 v = ((const float4*)(h + (size_t)s * 64))[q];
        float* o = out + (size_t)d * 64 + q * 4;
        atomicAdd(o + 0, coef * v.x);
        atomicAdd(o + 1, coef * v.y);
        atomicAdd(o + 2, coef * v.z);
        atomicAdd(o + 3, coef * v.w);
    }
}

// ---------------------------------------------------------------------------
extern "C" void kernel_launch(void* const* d_in, const int* in_sizes, int n_in,
                              void* d_out, int out_size, void* d_ws, size_t ws_size,
                              hipStream_t stream) {
    (void)n_in; (void)out_size; (void)d_ws; (void)ws_size;

    const float* h     = (const float*)d_in[0];
    const float* ini_h = (const float*)d_in[1];
    const float* bef_A = (const float*)d_in[2];
    const float* aft_A = (const float*)d_in[3];
    const float* bef_X = (const float*)d_in[4];
    const int*   src   = (const int*)d_in[5];
    const int*   dst   = (const int*)d_in[6];
    float*       out   = (float*)d_out;

    const int n_nodes = in_sizes[2];            // bef_A is [N,1]
    const int n_edges = in_sizes[5];            // src is [E]

    // Kernel 1: one wave per 16 nodes (8 waves / 256-thread block on wave32).
    const int wavesA  = (n_nodes + 15) / 16;
    const int blocksA = (wavesA * 32 + 255) / 256;
    hipLaunchKernelGGL(vsgc_init_wmma, dim3(blocksA), dim3(256), 0, stream,
                       h, ini_h, bef_X, out, n_nodes);

    // Kernel 2: grid-stride over edges, 2 edges per wave per iteration.
    const int blocksB = 2048;                   // 16384 waves -> ~24 edges/half-wave
    hipLaunchKernelGGL(vsgc_edge_scatter, dim3(blocksB), dim3(256), 0, stream,
                       h, aft_A, bef_A, src, dst, out, n_edges);
}